// LocallyConnectedLayer_1056561955143
// MI455X (gfx1250) — compile-verified
//
#include <hip/hip_runtime.h>
#include <cstdint>

// ---------------- problem constants ----------------
#define B_SZ   32          // batch
#define C_SZ   32          // in channels
#define O_SZ   64          // out channels
#define H_SZ   64
#define W_SZ   64
#define OH     62
#define OW     62
#define OHW    (OH * OW)   // 3844
#define KTOT   (C_SZ * 9)  // 288 = reduction length (c,i,j)

// ---------------- tiling ----------------
#define NW      8          // output-w locations per workgroup (store coalescing)
#define SA      292        // padded LDS row stride in floats (288 + 4 bank skew)
#define THREADS 256        // 8 wave32s: 2 (batch tiles) x 4 (o tiles)

typedef float v2f __attribute__((ext_vector_type(2)));
typedef float v8f __attribute__((ext_vector_type(8)));

__device__ __forceinline__ unsigned lds_off(const void* p) {
  // generic pointer to __shared__: low 32 bits hold the LDS byte offset
  return (unsigned)(uintptr_t)p;
}

// GLOBAL_LOAD_ASYNC_TO_LDS_B32 (GV mode, ASYNCcnt):
//   copies one dword from global vaddr to LDS vdst   (OFF applies to both sides)
template<int OFF>
__device__ __forceinline__ void async_b32(unsigned lds, const float* gp) {
  asm volatile("global_load_async_to_lds_b32 %0, %1, off offset:%c2"
               :: "v"(lds), "v"(gp), "n"(OFF) : "memory");
}

__device__ __forceinline__ void wait_async0() {
  asm volatile("s_wait_asynccnt 0x0" ::: "memory");
}

// Issue all async global->LDS loads for one (h,w) location.
//  A (x patches, 32 x 288):  thread t -> b = t>>3, channels c0..c0+3
//  B (W slice stored as Wl[o][k], 64 x 288): thread t -> o = t>>2, channels c0..c0+7
__device__ __forceinline__ void issue_loads(int t, int h, int w,
                                            float* Ab, float* Bb,
                                            const float* __restrict__ xg,
                                            const float* __restrict__ Wg) {
  {
    const int b  = t >> 3;
    const int c0 = (t & 7) * 4;
    for (int cc = 0; cc < 4; ++cc) {
      const int c = c0 + cc;
      const unsigned dst0 = lds_off(Ab + b * SA + c * 9);
      const float*   g0   = xg + (((size_t)(b * C_SZ + c) * H_SZ + h) * W_SZ + w);
      #pragma unroll
      for (int i = 0; i < 3; ++i) {                 // patch rows
        const unsigned d = dst0 + (unsigned)(12 * i);
        const float*   g = g0 + i * W_SZ;
        async_b32<0>(d, g); async_b32<4>(d, g); async_b32<8>(d, g);  // j = 0..2
      }
    }
  }
  {
    const int o  = t >> 2;
    const int c0 = (t & 3) * 8;
    const int hw = h * OW + w;
    for (int cc = 0; cc < 8; ++cc) {
      const int c = c0 + cc;
      const unsigned dst = lds_off(Bb + o * SA + c * 9);
      // W[o][c][h][w][i][j]: the 9 (i,j) taps are contiguous in memory
      const float* g = Wg + ((size_t)(o * C_SZ + c) * OHW + hw) * 9;
      async_b32<0>(dst, g);  async_b32<4>(dst, g);  async_b32<8>(dst, g);
      async_b32<12>(dst, g); async_b32<16>(dst, g); async_b32<20>(dst, g);
      async_b32<24>(dst, g); async_b32<28>(dst, g); async_b32<32>(dst, g);
    }
  }
}

// One 16x16 output tile per wave; K=288 via 72 x V_WMMA_F32_16X16X4_F32.
// f32 A 16x4 layout: lanes 0-15 hold K=0,1 (v0,v1); lanes 16-31 hold K=2,3.
// f32 B 4x16 mirrors it per column, so both fragments are float2 LDS loads.
__device__ __forceinline__ void compute_tile(int t, int wloc,
                                             const float* Ab, const float* Bb,
                                             float* stage) {
  const int lane = t & 31;
  const int wave = t >> 5;
  const int mi   = wave & 1;        // batch tile 0..1
  const int ni   = wave >> 1;       // out-channel tile 0..3
  const int r16  = lane & 15;
  const int ksel = (lane < 16) ? 0 : 2;
  const float* arow = Ab + (mi * 16 + r16) * SA + ksel;
  const float* brow = Bb + (ni * 16 + r16) * SA + ksel;

  v8f acc = {0.f, 0.f, 0.f, 0.f, 0.f, 0.f, 0.f, 0.f};
  #pragma unroll 4
  for (int k0 = 0; k0 < KTOT; k0 += 4) {
    v2f av = *(const v2f*)(arow + k0);   // ds_load_b64, 8B aligned
    v2f bv = *(const v2f*)(brow + k0);
    acc = __builtin_amdgcn_wmma_f32_16x16x4_f32(false, av, false, bv,
                                                (short)0, acc, false, false);
  }

  // C/D layout: vgpr r -> lanes 0-15: (M=r, N=lane); lanes 16-31: (M=r+8, N=lane-16)
  const int Mbase = mi * 16 + ((lane < 16) ? 0 : 8);
  const int oo    = ni * 16 + r16;
  #pragma unroll
  for (int r = 0; r < 8; ++r) {
    const int bb = Mbase + r;
    stage[(bb * O_SZ + oo) * NW + wloc] = acc[r];
  }
}

extern "C" __global__ __launch_bounds__(THREADS)
void lc_wmma_kernel(const float* __restrict__ xg, const float* __restrict__ Wg,
                    const float* __restrict__ bias, float* __restrict__ out) {
  extern __shared__ float smem[];
  float* Ab0   = smem;                    // 32*292 floats
  float* Bb0   = Ab0 + B_SZ * SA;         // 64*292 floats
  float* Ab1   = Bb0 + O_SZ * SA;
  float* Bb1   = Ab1 + B_SZ * SA;
  float* stage = Bb1 + O_SZ * SA;         // 2048 * NW floats

  const int t  = (int)threadIdx.x;
  const int h  = (int)blockIdx.x;
  const int w0 = (int)blockIdx.y * NW;
  const int nw = (OW - w0 < NW) ? (OW - w0) : NW;

  // double-buffered pipeline: async loads for loc l+1 overlap GEMM of loc l
  issue_loads(t, h, w0, Ab0, Bb0, xg, Wg);
  for (int l = 0; l < nw; ++l) {
    wait_async0();          // this wave's loads for location l have landed in LDS
    __syncthreads();        // all waves' loads have landed
    if (l + 1 < nw) {
      float* An = ((l + 1) & 1) ? Ab1 : Ab0;
      float* Bn = ((l + 1) & 1) ? Bb1 : Bb0;
      issue_loads(t, h, w0 + l + 1, An, Bn, xg, Wg);
    }
    const float* Ac = (l & 1) ? Ab1 : Ab0;
    const float* Bc = (l & 1) ? Bb1 : Bb0;
    compute_tile(t, l, Ac, Bc, stage);
    __syncthreads();        // buffer free for reuse; staging visible
  }

  // flush staging: lanes 0..7 write 8 consecutive w (32B contiguous per (b,o))
  for (int g = 0; g < (B_SZ * O_SZ * NW) / THREADS; ++g) {
    const int e    = t + g * THREADS;
    const int wloc = e & (NW - 1);
    const int bo   = e >> 3;            // b*64 + o
    if (wloc < nw) {
      const int oo  = bo & (O_SZ - 1);
      const int col = h * OW + w0 + wloc;
      out[(size_t)bo * OHW + col] = stage[bo * NW + wloc]
                                  + bias[(size_t)oo * OHW + col];
    }
  }
}

extern "C" void kernel_launch(void* const* d_in, const int* in_sizes, int n_in,
                              void* d_out, int out_size, void* d_ws, size_t ws_size,
                              hipStream_t stream) {
  const float* x    = (const float*)d_in[0];
  const float* Wg   = (const float*)d_in[1];
  const float* bias = (const float*)d_in[2];
  float* out = (float*)d_out;

  const size_t shmem =
      (size_t)(2 * (B_SZ + O_SZ) * SA + B_SZ * O_SZ * NW) * sizeof(float); // 283 KB
  (void)hipFuncSetAttribute((const void*)lc_wmma_kernel,
                            hipFuncAttributeMaxDynamicSharedMemorySize, (int)shmem);

  dim3 grid(OH, (OW + NW - 1) / NW);   // 62 x 8 workgroups
  lc_wmma_kernel<<<grid, dim3(THREADS), shmem, stream>>>(x, Wg, bias, out);
}